// GraphSageNet_64518998721150
// MI455X (gfx1250) — compile-verified
//
#include <hip/hip_runtime.h>

#define N_NODES 100000
#define N_EDGES 1600000
#define DIM 128
#define NLAYERS 4
#define NCLS 10

typedef __attribute__((ext_vector_type(2))) float v2f;
typedef __attribute__((ext_vector_type(8))) float v8f;
typedef __attribute__((ext_vector_type(4))) unsigned int u32x4;
typedef __attribute__((ext_vector_type(4))) int i32x4;
typedef __attribute__((ext_vector_type(8))) int i32x8;

#if defined(__has_builtin)
#if __has_builtin(__builtin_amdgcn_tensor_load_to_lds) && __has_builtin(__builtin_amdgcn_s_wait_tensorcnt)
#define USE_TDM 1
#endif
#endif

// D = A(16x4 f32) * B(4x16 f32) + C(16x16 f32), wave32 WMMA
__device__ __forceinline__ v8f wmma4(v2f a, v2f b, v8f c) {
  return __builtin_amdgcn_wmma_f32_16x16x4_f32(
      /*neg_a=*/false, a, /*neg_b=*/false, b,
      /*c_mod=*/(short)0, c, /*reuse_a=*/false, /*reuse_b=*/false);
}

#if USE_TDM
// Tensor Data Mover: DMA a 128x128 f32 tile (row stride = `stride` elems)
// from global memory into LDS at `lds`. Issued by one wave; tracked by TENSORcnt.
__device__ __forceinline__ void tdm_load_tile_128x128(const float* gptr, float* lds,
                                                      unsigned int stride_elems) {
  unsigned long long ga = (unsigned long long)(size_t)gptr;
  unsigned int ldsoff = (unsigned int)(size_t)(void*)lds;  // addr[31:0] == LDS byte offset
  u32x4 g0;
  g0[0] = 1u;                                    // count=1 (valid user descriptor)
  g0[1] = ldsoff;                                // lds_addr (bytes)
  g0[2] = (unsigned int)(ga & 0xFFFFFFFFu);      // global_addr[31:0]
  g0[3] = (unsigned int)((ga >> 32) & 0x1FFFFFFu) | (2u << 30);  // global_addr[56:32] | type=2
  i32x8 g1;
  g1[0] = (int)(2u << 16);            // workgroup_mask=0 | data_size=2 (4 bytes)
  g1[1] = (int)(stride_elems << 16);  // tensor_dim0 (bits 79:48) = row length
  g1[2] = (int)(128u << 16);          // tensor_dim1 = 128 (bits 111:80)
  g1[3] = (int)(128u << 16);          // tile_dim0   = 128 (bits 127:112)
  g1[4] = 128;                        // tile_dim1   = 128 (bits 143:128), tile_dim2 = 0
  g1[5] = (int)stride_elems;          // tensor_dim0_stride[31:0]
  g1[6] = 0;                          // stride hi / tensor_dim1_stride lo
  g1[7] = 0;                          // tensor_dim1_stride hi
  i32x4 z4 = {0, 0, 0, 0};
#if __clang_major__ >= 23
  i32x8 z8 = {0, 0, 0, 0, 0, 0, 0, 0};
  __builtin_amdgcn_tensor_load_to_lds(g0, g1, z4, z4, z8, 0);
#else
  __builtin_amdgcn_tensor_load_to_lds(g0, g1, z4, z4, 0);
#endif
}
#endif

// ---------------- one-time weight transpose: WT[n*K + k] = W[k*128 + n] ------
__global__ void transpose_kernel(const float* __restrict__ W, float* __restrict__ WT,
                                 int K) {
  int o = blockIdx.x * blockDim.x + threadIdx.x;  // output index, coalesced writes
  if (o < K * DIM) {
    int n = o / K;
    int k = o - n * K;
    WT[o] = W[(size_t)k * DIM + n];
  }
}

// ---------------- degree / normalization ----------------
__global__ void deg_count_kernel(const int* __restrict__ dst, float* __restrict__ deg) {
  int e = blockIdx.x * blockDim.x + threadIdx.x;
  if (e < N_EDGES) atomicAdd(&deg[dst[e]], 1.0f);
}

__global__ void norm_kernel(const float* __restrict__ deg,
                            float* __restrict__ nrm, float* __restrict__ ascl) {
  int i = blockIdx.x * blockDim.x + threadIdx.x;
  if (i < N_NODES) {
    float d = fmaxf(deg[i], 1.0f);
    float nm = rsqrtf(d);
    nrm[i]  = nm;        // deg^-0.5
    ascl[i] = nm / d;    // inv_deg * norm (applied to agg rows)
  }
}

// ---------------- edge gather + scatter-add ----------------
// 32 lanes per edge, float4 per lane, fp32 atomics into agg (L2-side).
__global__ void edge_scatter_kernel(const float* __restrict__ h,
                                    const float* __restrict__ nrm,
                                    const int* __restrict__ src,
                                    const int* __restrict__ dst,
                                    float* __restrict__ agg) {
  long long tid = (long long)blockIdx.x * blockDim.x + threadIdx.x;
  int lane = (int)(tid & 31);
  long long e = tid >> 5;
  if (e >= N_EDGES) return;
  int s = src[e];
  int t = dst[e];
  // pull a future source row toward L2 (h fits in the 192MB L2)
  if (lane == 0 && e + 4096 < N_EDGES) {
    __builtin_prefetch(h + (size_t)src[e + 4096] * DIM, 0, 1);
  }
  float ns = nrm[s];
  const float4 v = *(const float4*)(h + (long long)s * DIM + lane * 4);
  float* a = agg + (long long)t * DIM + lane * 4;
  atomicAdd(a + 0, v.x * ns);
  atomicAdd(a + 1, v.y * ns);
  atomicAdd(a + 2, v.z * ns);
  atomicAdd(a + 3, v.w * ns);
}

// ---------------- WMMA GEMM: C[N x 128] = act(A[N x KDIM] @ B + bias) (+resid) --
// BT is B pre-transposed: BT[n*KDIM + k]. TDM streams 128x128 tiles of BT into
// LDS (row-major == the transposed layout), giving contiguous b64 fragments.
// HALF2: A = [A0 | A1 * rowscale]  (the GraphSAGE concat, assembled on the fly)
template <int KDIM, bool RELU, bool RESID, bool HALF2>
__global__ __launch_bounds__(256) void gemm_kernel(
    const float* __restrict__ A0, const float* __restrict__ A1,
    const float* __restrict__ rowscale,
    const float* __restrict__ BT, const float* __restrict__ bias,
    const float* __restrict__ resid, float* __restrict__ C) {
  __shared__ float Bs[DIM * DIM];  // 64KB: Bs[n*128 + k] (transposed tile)

  const int lane = threadIdx.x & 31;
  const int wave = threadIdx.x >> 5;
  const int row0 = blockIdx.x * 128 + wave * 16;
  const int mloc  = lane & 15;        // M within tile (A frag) / N within tile (B frag)
  const int khalf = (lane >> 4) * 2;  // K-pair select per ISA layout

  int mrow = row0 + mloc;
  if (mrow >= N_NODES) mrow = N_NODES - 1;  // clamp; stores are guarded
  float sc = 1.0f;
  if (HALF2) sc = rowscale[mrow];

  v8f zero = {};
  v8f acc[8];
#pragma unroll
  for (int i = 0; i < 8; ++i) acc[i] = zero;

  for (int kb0 = 0; kb0 < KDIM; kb0 += DIM) {
    // stage BT columns [kb0, kb0+128) into LDS
#if USE_TDM
    if (threadIdx.x < 32) {
      tdm_load_tile_128x128(BT + kb0, Bs, (unsigned int)KDIM);
      __builtin_amdgcn_s_wait_tensorcnt(0);
    }
    __syncthreads();
#else
    for (int i = threadIdx.x; i < DIM * DIM; i += 256) {
      int n = i >> 7;
      int kl = i & 127;
      Bs[i] = BT[(size_t)n * KDIM + kb0 + kl];
    }
    __syncthreads();
#endif

    const bool second = HALF2 && (kb0 > 0);
    const float* arow = (second ? A1 : A0) + (size_t)mrow * DIM;

#pragma unroll 4
    for (int k0 = 0; k0 < DIM; k0 += 4) {
      v2f a = *(const v2f*)(arow + k0 + khalf);
      if (second) { a.x *= sc; a.y *= sc; }
#pragma unroll
      for (int nt = 0; nt < 8; ++nt) {
        int n = nt * 16 + mloc;
        v2f b = *(const v2f*)(&Bs[n * DIM + k0 + khalf]);  // contiguous -> ds_load b64
        acc[nt] = wmma4(a, b, acc[nt]);
      }
    }
    __syncthreads();
  }

  // epilogue: C/D layout -> lane%16 = N, VGPR v -> M = v + 8*(lane/16)
  const int nlane = lane & 15;
  const int mbase = (lane >> 4) * 8;
#pragma unroll
  for (int nt = 0; nt < 8; ++nt) {
    int n = nt * 16 + nlane;
    float bv = bias[n];
#pragma unroll
    for (int v = 0; v < 8; ++v) {
      int row = row0 + mbase + v;
      if (row < N_NODES) {
        float x = acc[nt][v] + bv;
        if (RELU) x = fmaxf(x, 0.0f);
        if (RESID) x += resid[(size_t)row * DIM + n];
        C[(size_t)row * DIM + n] = x;
      }
    }
  }
}

// ---------------- mean pooling + readout MLP ----------------
__global__ void pool_kernel(const float* __restrict__ h, float* __restrict__ hg) {
  int col = threadIdx.x;  // 128 threads
  int r0 = blockIdx.x * 256;
  int rend = r0 + 256;
  if (rend > N_NODES) rend = N_NODES;
  float s = 0.0f;
  for (int r = r0; r < rend; ++r) s += h[(size_t)r * DIM + col];
  atomicAdd(&hg[col], s);
}

__global__ void readout_kernel(const float* __restrict__ hg,
                               const float* __restrict__ rW1, const float* __restrict__ rb1,
                               const float* __restrict__ rW2, const float* __restrict__ rb2,
                               const float* __restrict__ rW3, const float* __restrict__ rb3,
                               float* __restrict__ out) {
  __shared__ float x[DIM];
  __shared__ float x1[64];
  __shared__ float x2[32];
  int t = threadIdx.x;  // 128 threads
  x[t] = hg[t] * (1.0f / (float)N_NODES);
  __syncthreads();
  if (t < 64) {
    float s = rb1[t];
    for (int k = 0; k < DIM; ++k) s += x[k] * rW1[k * 64 + t];
    x1[t] = fmaxf(s, 0.0f);
  }
  __syncthreads();
  if (t < 32) {
    float s = rb2[t];
    for (int k = 0; k < 64; ++k) s += x1[k] * rW2[k * 32 + t];
    x2[t] = fmaxf(s, 0.0f);
  }
  __syncthreads();
  if (t < NCLS) {
    float s = rb3[t];
    for (int k = 0; k < 32; ++k) s += x2[k] * rW3[k * NCLS + t];
    out[t] = s;
  }
}

extern "C" void kernel_launch(void* const* d_in, const int* in_sizes, int n_in,
                              void* d_out, int out_size, void* d_ws, size_t ws_size,
                              hipStream_t stream) {
  const float* h_in  = (const float*)d_in[0];
  const int*   src   = (const int*)d_in[1];
  const int*   dst   = (const int*)d_in[2];
  const float* enc_W = (const float*)d_in[3];
  const float* enc_b = (const float*)d_in[4];
  const float* W1    = (const float*)d_in[5];
  const float* b1    = (const float*)d_in[6];
  const float* W2    = (const float*)d_in[7];
  const float* b2    = (const float*)d_in[8];
  const float* rW1   = (const float*)d_in[9];
  const float* rb1   = (const float*)d_in[10];
  const float* rW2   = (const float*)d_in[11];
  const float* rb2   = (const float*)d_in[12];
  const float* rW3   = (const float*)d_in[13];
  const float* rb3   = (const float*)d_in[14];
  float* out = (float*)d_out;

  char* ws = (char*)d_ws;
  size_t off = 0;
  float* h_cur = (float*)(ws + off); off += (size_t)N_NODES * DIM * sizeof(float);
  float* agg   = (float*)(ws + off); off += (size_t)N_NODES * DIM * sizeof(float);  // also z
  float* deg   = (float*)(ws + off); off += (size_t)N_NODES * sizeof(float);
  float* nrm   = (float*)(ws + off); off += (size_t)N_NODES * sizeof(float);
  float* ascl  = (float*)(ws + off); off += (size_t)N_NODES * sizeof(float);
  float* hg    = (float*)(ws + off); off += (size_t)DIM * sizeof(float);
  float* wt_enc = (float*)(ws + off); off += (size_t)DIM * DIM * sizeof(float);
  float* wt1    = (float*)(ws + off); off += (size_t)NLAYERS * 2 * DIM * DIM * sizeof(float);
  float* wt2    = (float*)(ws + off); off += (size_t)NLAYERS * DIM * DIM * sizeof(float);

  hipMemsetAsync(deg, 0, (size_t)N_NODES * sizeof(float), stream);
  hipMemsetAsync(hg, 0, (size_t)DIM * sizeof(float), stream);

  // one-time weight transposes (tiny; amortized over 782 blocks x 9 GEMMs)
  transpose_kernel<<<(DIM * DIM + 255) / 256, 256, 0, stream>>>(enc_W, wt_enc, DIM);
  for (int l = 0; l < NLAYERS; ++l) {
    transpose_kernel<<<(2 * DIM * DIM + 255) / 256, 256, 0, stream>>>(
        W1 + (size_t)l * 2 * DIM * DIM, wt1 + (size_t)l * 2 * DIM * DIM, 2 * DIM);
    transpose_kernel<<<(DIM * DIM + 255) / 256, 256, 0, stream>>>(
        W2 + (size_t)l * DIM * DIM, wt2 + (size_t)l * DIM * DIM, DIM);
  }

  deg_count_kernel<<<(N_EDGES + 255) / 256, 256, 0, stream>>>(dst, deg);
  norm_kernel<<<(N_NODES + 255) / 256, 256, 0, stream>>>(deg, nrm, ascl);

  const int gblocks = (N_NODES + 127) / 128;

  // encoder: h = h_in @ enc_W + enc_b
  gemm_kernel<128, false, false, false><<<gblocks, 256, 0, stream>>>(
      h_in, nullptr, nullptr, wt_enc, enc_b, nullptr, h_cur);

  const long long ethreads = (long long)N_EDGES * 32;
  const int eblocks = (int)((ethreads + 255) / 256);

  for (int l = 0; l < NLAYERS; ++l) {
    hipMemsetAsync(agg, 0, (size_t)N_NODES * DIM * sizeof(float), stream);
    edge_scatter_kernel<<<eblocks, 256, 0, stream>>>(h_cur, nrm, src, dst, agg);
    // z = relu([h | agg*inv_deg*norm] @ W1 + b1)   (z aliases agg: rows read before written)
    gemm_kernel<256, true, false, true><<<gblocks, 256, 0, stream>>>(
        h_cur, agg, ascl, wt1 + (size_t)l * 2 * DIM * DIM, b1 + (size_t)l * DIM,
        nullptr, agg);
    // h = h + relu(z @ W2 + b2)   (in-place residual)
    gemm_kernel<128, true, true, false><<<gblocks, 256, 0, stream>>>(
        agg, nullptr, nullptr, wt2 + (size_t)l * DIM * DIM, b2 + (size_t)l * DIM,
        h_cur, h_cur);
  }

  pool_kernel<<<(N_NODES + 255) / 256, 128, 0, stream>>>(h_cur, hg);
  readout_kernel<<<1, 128, 0, stream>>>(hg, rW1, rb1, rW2, rb2, rW3, rb3, out);
}